// AgentNetwork_70514773066176
// MI455X (gfx1250) — compile-verified
//
#include <hip/hip_runtime.h>
#include <cstdint>

// ---------------------------------------------------------------------------
// AgentNetwork forward for MI455X (gfx1250, wave32).
// HBM-bound GEMV chain (~91 MB weights -> ~4us floor @ 23.3 TB/s).
// Big GEMVs: async-double-buffered LDS staging + WMMA f32 16x16x4 dot engine.
// ---------------------------------------------------------------------------

#define HDIM 2048
#define ADIM 8
#define VDIM 16
#define PDIM 256

#define CK    256          // K-chunk staged to LDS per iteration
#define LDSP  (CK + 4)     // padded LDS pitch: 16B-aligned rows, bank-spread
#define NWAVE 8
#define RPB   16           // output rows per block (one WMMA M-tile)

typedef __attribute__((ext_vector_type(2))) float v2f;
typedef __attribute__((ext_vector_type(4))) float v4f;
typedef __attribute__((ext_vector_type(8))) float v8f;
typedef __attribute__((ext_vector_type(4))) int   v4i;

// Pointer-to-int4 in global (AS1) / shared (AS3) address spaces, as expected
// by the async-to-LDS builtin (diagnostic showed `int __vector(4) __device__*`).
typedef __attribute__((address_space(1))) v4i gas_v4i;
typedef __attribute__((address_space(3))) v4i las_v4i;

#if defined(__AMDGCN__) && __has_builtin(__builtin_amdgcn_global_load_async_to_lds_b128)
#define HAVE_ASYNC_LDS 1
#endif

// Async global->LDS 16-byte copy (ASYNCcnt-tracked) with sync fallback.
__device__ __forceinline__ void stage_b128(const float* src, float* dst) {
#ifdef HAVE_ASYNC_LDS
    __builtin_amdgcn_global_load_async_to_lds_b128(
        (gas_v4i*)(unsigned long long)(uintptr_t)src,
        (las_v4i*)(unsigned int)(uintptr_t)dst,
        0, 0);
#else
    *(v4f*)dst = *(const v4f*)src;
#endif
}

__device__ __forceinline__ void wait_async_all() {
#ifdef HAVE_ASYNC_LDS
#if __has_builtin(__builtin_amdgcn_s_wait_asynccnt)
    __builtin_amdgcn_s_wait_asynccnt(0);
#else
    asm volatile("s_wait_asynccnt 0x0" ::: "memory");
#endif
#endif
}

// ---------------------------------------------------------------------------
// Threefry-2x32 based gumbel noise (categorical sampling = argmax(l + g))
// ---------------------------------------------------------------------------
__device__ __forceinline__ unsigned rotl32(unsigned x, int r) {
    return (x << r) | (x >> (32 - r));
}

__device__ __forceinline__ void threefry2x32(unsigned k0, unsigned k1,
                                             unsigned c0, unsigned c1,
                                             unsigned& o0, unsigned& o1) {
    const unsigned R0[4] = {13, 15, 26, 6};
    const unsigned R1[4] = {17, 29, 16, 24};
    unsigned ks[3] = {k0, k1, 0x1BD11BDAu ^ k0 ^ k1};
    unsigned x0 = c0 + ks[0];
    unsigned x1 = c1 + ks[1];
#pragma unroll
    for (int g = 0; g < 5; ++g) {
        const unsigned* R = (g & 1) ? R1 : R0;
#pragma unroll
        for (int r = 0; r < 4; ++r) {
            x0 += x1;
            x1 = rotl32(x1, R[r]);
            x1 ^= x0;
        }
        unsigned s = (unsigned)(g + 1);
        x0 += ks[s % 3];
        x1 += ks[(s + 1) % 3] + s;
    }
    o0 = x0;
    o1 = x1;
}

__device__ __forceinline__ float gumbel_noise(unsigned head, unsigned i) {
    unsigned r0, r1;
    threefry2x32(42u, 0u, head, i, r0, r1);
    float u = ((float)(r0 >> 8) + 0.5f) * (1.0f / 16777216.0f);  // (0,1)
    return -__logf(-__logf(u));
}

// ---------------------------------------------------------------------------
// Big GEMV: out[row] = relu( W[sel][row, :2048] @ x
//                          + sum_j W[sel][row, off_j + samples[j]]
//                          + b[sel][row] )
// Grid: HDIM/RPB = 128 blocks, 256 threads (8 waves).
// Each block owns 16 rows; waves split K; WMMA f32 16x16x4 accumulates.
// Double-buffered async global->LDS staging overlaps the next 16 KB W-tile
// with the current tile's WMMAs.
// ---------------------------------------------------------------------------
__global__ __launch_bounds__(256) void gemv_wmma_relu(
    const float* __restrict__ Wbase, const float* __restrict__ Bbase,
    const int* __restrict__ sel, long long selStrideW, long long selStrideB,
    int inPitch, const float* __restrict__ x,
    const int* __restrict__ samples, int n_extra,
    float* __restrict__ out) {
    __shared__ float xs[HDIM];
    __shared__ float ws[2][RPB * LDSP];
    __shared__ float red[NWAVE][RPB];

    const int tid  = threadIdx.x;
    const int lane = tid & 31;
    const int wave = tid >> 5;
    const int rowBase = blockIdx.x * RPB;

    const int s = sel ? sel[0] : 0;
    const float* W  = Wbase + (long long)s * selStrideW;
    const float* Bv = Bbase + (long long)s * selStrideB;

    // Per-thread staging slice: 16 threads/row, 16 consecutive floats each.
    const int r  = tid >> 4;
    const int ci = tid & 15;
    const float* wrow = W + (long long)(rowBase + r) * inPitch + ci * 16;
    float* dst0 = &ws[0][r * LDSP + ci * 16];
    float* dst1 = &ws[1][r * LDSP + ci * 16];

    // Stage x (8 KB, one-time) and W chunk 0 into buffer 0.
    stage_b128(x + tid * 4,        xs + tid * 4);
    stage_b128(x + 1024 + tid * 4, xs + 1024 + tid * 4);
#pragma unroll
    for (int q = 0; q < 4; ++q) stage_b128(wrow + q * 4, dst0 + q * 4);
    wait_async_all();
    __syncthreads();

    v8f acc = {};

    for (int c0 = 0; c0 < HDIM; c0 += CK) {
        const int buf = (c0 / CK) & 1;
        // Prefetch next chunk into the other buffer (overlaps WMMA below).
        if (c0 + CK < HDIM) {
            const float* src = wrow + c0 + CK;
            float* dst = buf ? dst0 : dst1;
#pragma unroll
            for (int q = 0; q < 4; ++q) stage_b128(src + q * 4, dst + q * 4);
        }

        // Wave w consumes K slice [w*32, w*32+32) of this chunk: 8 WMMAs.
        const float* wb = ws[buf];
        const int kw = wave * 32;
#pragma unroll
        for (int kk = 0; kk < 32; kk += 4) {
            const int koff = kw + kk + 2 * (lane >> 4);
            // A (16x4 f32): lane l -> row (l&15), K pair {koff, koff+1}
            v2f a = *(const v2f*)(wb + (lane & 15) * LDSP + koff);
            // B (4x16 f32): x chunk replicated across all 16 N columns
            v2f b = *(const v2f*)(xs + c0 + koff);
            acc = __builtin_amdgcn_wmma_f32_16x16x4_f32(
                false, a, false, b, (short)0, acc, false, false);
        }

        wait_async_all();   // next chunk fully in LDS (this wave's asyncs)
        __syncthreads();    // ... and everyone else's
    }

    // D layout: vgpr m, lane 0 -> (M=m, N=0); lane 16 -> (M=m+8, N=0).
    if (lane == 0 || lane == 16) {
        const int mBase = (lane >> 4) * 8;
#pragma unroll
        for (int m = 0; m < 8; ++m)
            red[wave][mBase + m] = acc[m];
    }
    __syncthreads();

    if (wave == 0 && lane < RPB) {
        float v = 0.0f;
#pragma unroll
        for (int w = 0; w < NWAVE; ++w) v += red[w][lane];
        const int row = rowBase + lane;
        v += Bv[row];
        // One-hot concat parts: single-column contributions.
        const int offs[4] = {2048, 2056, 2072, 2328};
        for (int j = 0; j < n_extra; ++j)
            v += W[(long long)row * inPitch + offs[j] + samples[j]];
        out[row] = fmaxf(v, 0.0f);
    }
}

// ---------------------------------------------------------------------------
// Head: logits = W[sel] @ x + b[sel]; sample via gumbel-argmax; logprob via
// logsumexp. Optionally the critic (Wc[sel] @ x + bc[sel]) on wave 1.
// Single block, 256 threads.
// ---------------------------------------------------------------------------
__global__ __launch_bounds__(256) void head_sample(
    const float* __restrict__ Wbase, const float* __restrict__ bbase,
    const int* __restrict__ selp, long long wStride, long long bStride,
    int N, const float* __restrict__ x, int head,
    int* __restrict__ samples, float* __restrict__ outp,
    const float* __restrict__ WcBase, const float* __restrict__ bcBase) {
    __shared__ float lg[PDIM];

    const int tid  = threadIdx.x;
    const int lane = tid & 31;
    const int wave = tid >> 5;

    const int s = selp ? selp[0] : 0;
    const float* W = Wbase + (long long)s * wStride;
    const float* b = bbase + (long long)s * bStride;

    // One row per wave per iteration; lane-parallel, coalesced dot.
    for (int r = wave; r < N; r += NWAVE) {
        const float* wr = W + (long long)r * HDIM;
        float p = 0.0f;
        for (int k = lane; k < HDIM; k += 32) p += wr[k] * x[k];
#pragma unroll
        for (int off = 16; off > 0; off >>= 1)
            p += __shfl_down(p, off, 32);
        if (lane == 0) lg[r] = p + b[r];
    }
    __syncthreads();

    if (WcBase && wave == 1) {  // critic (only on last head)
        const float* wr = WcBase + (long long)s * HDIM;
        float p = 0.0f;
        for (int k = lane; k < HDIM; k += 32) p += wr[k] * x[k];
#pragma unroll
        for (int off = 16; off > 0; off >>= 1)
            p += __shfl_down(p, off, 32);
        if (lane == 0) outp[10] = p + bcBase[s];
    }

    if (wave == 0) {
        float best = -3.4e38f, mx = -3.4e38f;
        int bi = 0;
        for (int i = lane; i < N; i += 32) {
            const float li = lg[i];
            const float t  = li + gumbel_noise((unsigned)head, (unsigned)i);
            if (t > best) { best = t; bi = i; }
            mx = fmaxf(mx, li);
        }
#pragma unroll
        for (int off = 16; off > 0; off >>= 1) {
            const float ob = __shfl_down(best, off, 32);
            const int   oi = __shfl_down(bi, off, 32);
            if (ob > best) { best = ob; bi = oi; }
            mx = fmaxf(mx, __shfl_down(mx, off, 32));
        }
        mx = __shfl(mx, 0, 32);
        const int sidx = __shfl(bi, 0, 32);
        float se = 0.0f;
        for (int i = lane; i < N; i += 32) se += __expf(lg[i] - mx);
#pragma unroll
        for (int off = 16; off > 0; off >>= 1)
            se += __shfl_down(se, off, 32);
        if (lane == 0) {
            const float lse = mx + __logf(se);
            samples[head]  = sidx;
            outp[head]     = (float)sidx;       // sampled_values[head]
            outp[5 + head] = lg[sidx] - lse;    // log_probs[head]
        }
    }
}

// ---------------------------------------------------------------------------
// Launcher: 10 stream-ordered launches; sampled indices flow through d_ws so
// weight-slice selection (Wh*[a]) happens on device (graph-capture safe).
// ---------------------------------------------------------------------------
extern "C" void kernel_launch(void* const* d_in, const int* in_sizes, int n_in,
                              void* d_out, int out_size, void* d_ws, size_t ws_size,
                              hipStream_t stream) {
    (void)in_sizes; (void)n_in; (void)out_size; (void)ws_size;

    const float* state = (const float*)d_in[0];
    const float* Wa  = (const float*)d_in[1];  const float* ba  = (const float*)d_in[2];
    const float* Wap = (const float*)d_in[3];  const float* bap = (const float*)d_in[4];
    const float* Wv  = (const float*)d_in[5];  const float* bv  = (const float*)d_in[6];
    const float* Wvp = (const float*)d_in[7];  const float* bvp = (const float*)d_in[8];
    const float* Wh0 = (const float*)d_in[9];  const float* bh0 = (const float*)d_in[10];
    const float* Wp0 = (const float*)d_in[11]; const float* bp0 = (const float*)d_in[12];
    const float* Wh1 = (const float*)d_in[13]; const float* bh1 = (const float*)d_in[14];
    const float* Wp1 = (const float*)d_in[15]; const float* bp1 = (const float*)d_in[16];
    const float* Wh2 = (const float*)d_in[17]; const float* bh2 = (const float*)d_in[18];
    const float* Wp2 = (const float*)d_in[19]; const float* bp2 = (const float*)d_in[20];
    const float* Wc  = (const float*)d_in[21]; const float* bc  = (const float*)d_in[22];

    float* outf = (float*)d_out;
    float* wsf  = (float*)d_ws;
    float* hA   = wsf;                     // 2048 f32
    float* hB   = wsf + HDIM;              // 2048 f32
    int*   smp  = (int*)(wsf + 2 * HDIM);  // 5 sampled indices

    const int IN0 = HDIM + ADIM + VDIM;   // 2072
    const int IN1 = IN0 + PDIM;           // 2328
    const int IN2 = IN1 + PDIM;           // 2584

    const dim3 gG(HDIM / RPB);  // 128 blocks
    const dim3 gB(256);

    // action head
    gemv_wmma_relu<<<gG, gB, 0, stream>>>(Wa, ba, nullptr, 0, 0, HDIM,
                                          state, smp, 0, hA);
    head_sample<<<1, gB, 0, stream>>>(Wap, bap, nullptr, 0, 0, ADIM, hA, 0,
                                      smp, outf, nullptr, nullptr);
    // visualisation head
    gemv_wmma_relu<<<gG, gB, 0, stream>>>(Wv, bv, nullptr, 0, 0, HDIM + ADIM,
                                          hA, smp, 1, hB);
    head_sample<<<1, gB, 0, stream>>>(Wvp, bvp, nullptr, 0, 0, VDIM, hB, 1,
                                      smp, outf, nullptr, nullptr);
    // param 0 (weights selected by sampled action)
    gemv_wmma_relu<<<gG, gB, 0, stream>>>(Wh0, bh0, smp, (long long)HDIM * IN0,
                                          HDIM, IN0, hB, smp, 2, hA);
    head_sample<<<1, gB, 0, stream>>>(Wp0, bp0, smp, (long long)PDIM * HDIM,
                                      PDIM, PDIM, hA, 2, smp, outf,
                                      nullptr, nullptr);
    // param 1
    gemv_wmma_relu<<<gG, gB, 0, stream>>>(Wh1, bh1, smp, (long long)HDIM * IN1,
                                          HDIM, IN1, hA, smp, 3, hB);
    head_sample<<<1, gB, 0, stream>>>(Wp1, bp1, smp, (long long)PDIM * HDIM,
                                      PDIM, PDIM, hB, 3, smp, outf,
                                      nullptr, nullptr);
    // param 2 + critic
    gemv_wmma_relu<<<gG, gB, 0, stream>>>(Wh2, bh2, smp, (long long)HDIM * IN2,
                                          HDIM, IN2, hB, smp, 4, hA);
    head_sample<<<1, gB, 0, stream>>>(Wp2, bp2, smp, (long long)PDIM * HDIM,
                                      PDIM, PDIM, hA, 4, smp, outf, Wc, bc);
}